// NetGCN_9234179686414
// MI455X (gfx1250) — compile-verified
//
#include <hip/hip_runtime.h>
#include <math.h>

typedef __attribute__((ext_vector_type(2))) float v2f;
typedef __attribute__((ext_vector_type(8))) float v8f;

#define NFEATS 128
#define DIM 64
#define NGRAPHS 128

// ---------------- zero / relu helpers (grid-stride) ----------------
__global__ __launch_bounds__(256) void zero_kernel(float* __restrict__ p, size_t n) {
    size_t i = (size_t)blockIdx.x * 256u + threadIdx.x;
    size_t stride = (size_t)gridDim.x * 256u;
    for (; i < n; i += stride) p[i] = 0.0f;
}

__global__ __launch_bounds__(256) void relu_kernel(float* __restrict__ p, size_t n) {
    size_t i = (size_t)blockIdx.x * 256u + threadIdx.x;
    size_t stride = (size_t)gridDim.x * 256u;
    for (; i < n; i += stride) p[i] = fmaxf(p[i], 0.0f);
}

// ---------------- WMMA f32 GEMM: out[nrows,64] = in[nrows,K] @ W[K,64] ----------------
// One wave computes a 16-row x 64-col tile as 4 N-tiles of 16x16, K in steps of 4
// using V_WMMA_F32_16X16X4_F32 (fp32 matrix core, wave32).
//
// A layout (ISA 7.12.2, 32-bit A 16x4): lane L -> M = L%16; lanes 0-15 hold K={0,1}
// in v0/v1, lanes 16-31 hold K={2,3}.  B (4x16) mirrors: lane L -> N = L%16,
// lanes 0-15 hold rows K={0,1}, lanes 16-31 rows K={2,3}.
// C/D (16x16 f32): VGPR i -> M = i (lanes 0-15) / M = 8+i (lanes 16-31), N = L%16.
//
// W is staged into LDS PRE-SWIZZLED into B-fragment order so each lane fetches its
// v2f fragment with one bank-conflict-free ds_load_b64:
//   sW[(((g*4 + t)*32 + L)*2 + p] = W[(4g + (L>=16?2:0) + p)*64 + t*16 + (L&15)]
// (g = k-group = k0/4, t = n-tile, L = lane, p = element of the K-pair)
template<int K>
__global__ __launch_bounds__(256) void gemm64_wmma(const float* __restrict__ in,
                                                   const float* __restrict__ W,
                                                   float* __restrict__ out,
                                                   int nrows) {
    __shared__ alignas(16) float sW[K * 64];
    for (int i = threadIdx.x; i < K * 64; i += 256) {
        const int p = i & 1;
        const int L = (i >> 1) & 31;
        const int t = (i >> 6) & 3;
        const int g = i >> 8;
        const int kb = g * 4 + ((L >= 16) ? 2 : 0) + p;
        const int n  = t * 16 + (L & 15);
        sW[i] = W[kb * 64 + n];
    }
    __syncthreads();

    const int wave = threadIdx.x >> 5;
    const int lane = threadIdx.x & 31;
    const int tile = blockIdx.x * 8 + wave;   // 16-row tile handled by this wave
    const int m0   = tile * 16;
    if (m0 >= nrows) return;                  // wave-uniform: EXEC stays all-1s for WMMA

    const bool hi = (lane >= 16);
    const int  mr = lane & 15;                // M within tile
    const int  nc = lane & 15;                // N within 16-col tile (implicit in swizzle)
    (void)nc;
    const int  koff = hi ? 2 : 0;
    const float* arow = in + (size_t)(m0 + mr) * K + koff;

    v8f acc0 = {}, acc1 = {}, acc2 = {}, acc3 = {};

    for (int k0 = 0; k0 < K; k0 += 4) {
        // A fragment: K = k0+koff, k0+koff+1 (8B aligned -> one global_load_b64)
        v2f a = *(const v2f*)(arow + k0);
        // B fragments: one ds_load_b64 per n-tile, pre-swizzled layout
        const v2f* wb = (const v2f*)(sW + (k0 << 6)) + lane;   // group base + lane
        v2f b0 = wb[0];
        v2f b1 = wb[32];
        v2f b2 = wb[64];
        v2f b3 = wb[96];
        acc0 = __builtin_amdgcn_wmma_f32_16x16x4_f32(false, a, false, b0, (short)0, acc0, false, false);
        acc1 = __builtin_amdgcn_wmma_f32_16x16x4_f32(false, a, false, b1, (short)0, acc1, false, false);
        acc2 = __builtin_amdgcn_wmma_f32_16x16x4_f32(false, a, false, b2, (short)0, acc2, false, false);
        acc3 = __builtin_amdgcn_wmma_f32_16x16x4_f32(false, a, false, b3, (short)0, acc3, false, false);
    }

    const int rbase = m0 + (hi ? 8 : 0);
    float* obase = out + (size_t)rbase * 64 + (lane & 15);
#pragma unroll
    for (int i = 0; i < 8; ++i) {
        float* orow = obase + (size_t)i * 64;
        orow[0]  = acc0[i];
        orow[16] = acc1[i];
        orow[32] = acc2[i];
        orow[48] = acc3[i];
    }
}

// ---------------- edge scatter: agg[dst] += h[src], 64 feats/edge ----------------
// thread -> (edge, 4-feature chunk): float4 gather + 4 f32 atomics (L2-resident:
// both node-feature arrays are 51.2 MB < 192 MB L2).
__global__ __launch_bounds__(256) void scatter_kernel(const float* __restrict__ h,
                                                      const int* __restrict__ src,
                                                      const int* __restrict__ dst,
                                                      float* __restrict__ agg,
                                                      int nedges) {
    size_t tid = (size_t)blockIdx.x * 256u + threadIdx.x;
    int e = (int)(tid >> 4);
    if (e >= nedges) return;
    int f = (int)(tid & 15) * 4;
    int s = src[e];
    int d = dst[e];
    float4 v = *(const float4*)(h + (size_t)s * DIM + f);
    float* ap = agg + (size_t)d * DIM + f;
    atomicAdd(ap + 0, v.x);
    atomicAdd(ap + 1, v.y);
    atomicAdd(ap + 2, v.z);
    atomicAdd(ap + 3, v.w);
}

// ---------------- per-graph pooled sums + counts ----------------
__global__ __launch_bounds__(256) void pool_kernel(const float* __restrict__ x,
                                                   const int* __restrict__ gids,
                                                   float* __restrict__ gsum,
                                                   float* __restrict__ gcnt,
                                                   int nnodes) {
    size_t tid = (size_t)blockIdx.x * 256u + threadIdx.x;
    int n = (int)(tid >> 6);
    if (n >= nnodes) return;
    int f = (int)(tid & 63);
    int g = gids[n];
    atomicAdd(&gsum[(size_t)g * DIM + f], x[(size_t)n * DIM + f]);
    if (f == 0) atomicAdd(&gcnt[g], 1.0f);
}

// ---------------- head: sigmoid(mean @ Wfc) ----------------
__global__ __launch_bounds__(128) void head_kernel(const float* __restrict__ gsum,
                                                   const float* __restrict__ gcnt,
                                                   const float* __restrict__ Wfc,
                                                   float* __restrict__ out) {
    int g = threadIdx.x;   // 128 graphs
    float c = fmaxf(gcnt[g], 1.0f);
    float acc = 0.0f;
#pragma unroll 8
    for (int f = 0; f < DIM; ++f)
        acc += (gsum[(size_t)g * DIM + f] / c) * Wfc[f];
    out[g] = 1.0f / (1.0f + expf(-acc));
}

extern "C" void kernel_launch(void* const* d_in, const int* in_sizes, int n_in,
                              void* d_out, int out_size, void* d_ws, size_t ws_size,
                              hipStream_t stream) {
    const float* features = (const float*)d_in[0];
    const int*   src      = (const int*)d_in[1];
    const int*   dst      = (const int*)d_in[2];
    const int*   gids     = (const int*)d_in[3];
    const float* W1       = (const float*)d_in[4];
    const float* W2       = (const float*)d_in[5];
    const float* Wfc      = (const float*)d_in[6];
    float* out = (float*)d_out;

    const int n_nodes = in_sizes[0] / NFEATS;   // 200000
    const int n_edges = in_sizes[1];            // 3200000

    const size_t nodef = (size_t)n_nodes * DIM; // 12.8M floats
    float* bufA = (float*)d_ws;                 // transformed features  [N,64]
    float* bufB = bufA + nodef;                 // aggregation buffer    [N,64]
    float* gsum = bufB + nodef;                 // [128,64]
    float* gcnt = gsum + (size_t)NGRAPHS * DIM; // [128]

    const int tiles   = (n_nodes + 15) / 16;
    const int gblocks = (tiles + 7) / 8;
    const size_t sth  = (size_t)n_edges * 16;
    const int sblocks = (int)((sth + 255) / 256);
    const size_t pth  = (size_t)n_nodes * DIM;
    const int pblocks = (int)((pth + 255) / 256);

    // 1. zero agg buffer + graph sums + counts (contiguous region)
    zero_kernel<<<2048, 256, 0, stream>>>(bufB, nodef + (size_t)NGRAPHS * DIM + NGRAPHS);
    // 2. h1 = X @ W1
    gemm64_wmma<NFEATS><<<gblocks, 256, 0, stream>>>(features, W1, bufA, n_nodes);
    // 3. agg1[dst] += h1[src]
    scatter_kernel<<<sblocks, 256, 0, stream>>>(bufA, src, dst, bufB, n_edges);
    // 4. relu in place
    relu_kernel<<<2048, 256, 0, stream>>>(bufB, nodef);
    // 5. h2 = relu(agg1) @ W2
    gemm64_wmma<DIM><<<gblocks, 256, 0, stream>>>(bufB, W2, bufA, n_nodes);
    // 6. re-zero agg buffer, then second aggregation
    zero_kernel<<<2048, 256, 0, stream>>>(bufB, nodef);
    scatter_kernel<<<sblocks, 256, 0, stream>>>(bufA, src, dst, bufB, n_edges);
    // 7. per-graph sums + counts
    pool_kernel<<<pblocks, 256, 0, stream>>>(bufB, gids, gsum, gcnt, n_nodes);
    // 8. sigmoid(mean @ Wfc)
    head_kernel<<<1, NGRAPHS, 0, stream>>>(gsum, gcnt, Wfc, out);
}